// PolicyNet_21406117004078
// MI455X (gfx1250) — compile-verified
//
#include <hip/hip_runtime.h>
#include <math.h>

// ---------------------------------------------------------------------------
// PolicyNet fused scorer for gfx1250 (MI455X), fp32 WMMA path.
//   rel = emb[id]                         [A,128]  (HBM gather, ~51MB)
//   att = rel @ Q^T ; w = softmax_L(att)  [A,64]
//   q   = w @ Q                           [A,128]
//   h   = relu(q @ W1q^T + consth)        [A,128]  (consth = b1 + hist@W1h^T)
//   out = h @ W2^T + b2                   [A,128]
//   s   = sum(rel*out, -1); softmax_A(s)  [A]
// One wave32 handles a 32-action tile (two 16-row M-tiles) so each B fragment
// read from LDS feeds TWO v_wmma_f32_16x16x4_f32 ops (halves DS-per-FLOP).
// ---------------------------------------------------------------------------

typedef __attribute__((ext_vector_type(2))) float v2f;
typedef __attribute__((ext_vector_type(4))) float v4f;
typedef __attribute__((ext_vector_type(8))) float v8f;

#define A_N   100000
#define D_N   128
#define L_N   64
#define LDA   132                 // padded LDS row stride: 132%64=4 -> conflict-free
#define WAVES 4                   // waves per block (128 threads)
#define MROWS 32                  // actions per wave (two 16-row WMMA tiles)
#define TILES ((A_N + MROWS - 1) / MROWS)   // 3125, exact

// dynamic LDS layout (float offsets)
#define SM_QT    0
#define SM_W1Q   (SM_QT  + L_N * LDA)     //  8448
#define SM_W2    (SM_W1Q + D_N * LDA)     // 25344
#define SM_CH    (SM_W2  + D_N * LDA)     // 42240
#define SM_B2    (SM_CH  + 128)
#define SM_WAVE  (SM_B2  + 128)
#define SM_PER_WAVE (2 * MROWS * LDA)     // rel tile + stage buffer
#define SM_TOTAL_F (SM_WAVE + WAVES * SM_PER_WAVE)
#define SM_BYTES   (SM_TOTAL_F * 4)       // 305,152 B < 320 KB/WGP

__device__ __forceinline__ v8f wmma4(v2f a, v2f b, v8f c) {
    // D = A(16x4) * B(4x16) + C(16x16), full fp32
    return __builtin_amdgcn_wmma_f32_16x16x4_f32(false, a, false, b,
                                                 (short)0, c, false, false);
}

// per-wave LDS ordering fence (DS ops are in-order per wave; this pins the
// compiler and drains DScnt before dependent cross-lane reads)
__device__ __forceinline__ void lds_wave_sync() {
    __builtin_amdgcn_wave_barrier();
    asm volatile("s_wait_dscnt 0" ::: "memory");
    __builtin_amdgcn_wave_barrier();
}

__device__ __forceinline__ float half16_reduce_max(float v) {
    #pragma unroll
    for (int m = 1; m <= 8; m <<= 1) v = fmaxf(v, __shfl_xor(v, m, 32));
    return v;
}
__device__ __forceinline__ float half16_reduce_sum(float v) {
    #pragma unroll
    for (int m = 1; m <= 8; m <<= 1) v += __shfl_xor(v, m, 32);
    return v;
}

// ---------------------------------------------------------------------------
// consth[i] = b1[i] + sum_d hist[d] * W1[i][d]   (hist half of the MLP input)
// ---------------------------------------------------------------------------
__global__ void consth_kernel(const float* __restrict__ hist,
                              const float* __restrict__ W1,
                              const float* __restrict__ b1,
                              float* __restrict__ consth) {
    int i = blockIdx.x * blockDim.x + threadIdx.x;
    if (i < D_N) {
        float acc = b1[i];
        #pragma unroll 4
        for (int d = 0; d < D_N; ++d) acc += hist[d] * W1[i * (2 * D_N) + d];
        consth[i] = acc;
    }
}

// ---------------------------------------------------------------------------
// Main fused kernel: 32 actions per wave, 4 waves per block.
// ---------------------------------------------------------------------------
__global__ __launch_bounds__(128)
void policy_scores_kernel(const int*   __restrict__ action_ids,
                          const float* __restrict__ emb_table,
                          const float* __restrict__ question_t,
                          const float* __restrict__ W1,
                          const float* __restrict__ W2,
                          const float* __restrict__ b2,
                          const float* __restrict__ consth,
                          float*       __restrict__ scores) {
    extern __shared__ float sm[];
    const int tid = threadIdx.x;

    // ---- cooperative load of block-shared operands -------------------------
    for (int idx = tid; idx < L_N * 32; idx += 128) {       // question_t [64][128]
        int r = idx >> 5, c = (idx & 31) * 4;
        *(v4f*)(sm + SM_QT + r * LDA + c) = *(const v4f*)(question_t + r * D_N + c);
    }
    for (int idx = tid; idx < D_N * 32; idx += 128) {       // W1q [128][128], W2 [128][128]
        int r = idx >> 5, c = (idx & 31) * 4;
        *(v4f*)(sm + SM_W1Q + r * LDA + c) = *(const v4f*)(W1 + r * (2 * D_N) + D_N + c);
        *(v4f*)(sm + SM_W2  + r * LDA + c) = *(const v4f*)(W2 + r * D_N + c);
    }
    if (tid < 128) {
        sm[SM_CH + tid] = consth[tid];
        sm[SM_B2 + tid] = b2[tid];
    }
    __syncthreads();

    const int wave = tid >> 5;
    const int lane = tid & 31;
    const int half = lane >> 4;    // K-half for A/B fragments
    const int l16  = lane & 15;    // M (A) / N (B,D) lane index

    int tile = blockIdx.x * WAVES + wave;
    if (tile >= TILES) tile = TILES - 1;   // tail waves redo last tile (same values)
    const int a0 = tile * MROWS;

    float* s_rel = sm + SM_WAVE + wave * SM_PER_WAVE;  // [32][LDA]
    float* s_stg = s_rel + MROWS * LDA;                // [32][LDA] staging

    // ---- gather 32 relation embeddings (HBM-bound part) --------------------
    #pragma unroll 4
    for (int r = 0; r < MROWS; ++r) {
        int id = action_ids[a0 + r];
        v4f v = *(const v4f*)(emb_table + (size_t)id * D_N + lane * 4);
        *(v4f*)(s_rel + r * LDA + lane * 4) = v;
    }
    lds_wave_sync();

    // ---- GEMM1: att[32x64] = rel[32x128] * Q^T  (B^T == Q row-major) -------
    v8f att[2][4] = {};
    for (int kb = 0; kb < 32; ++kb) {
        v2f a0f = *(const v2f*)(s_rel + (     l16) * LDA + 4 * kb + 2 * half);
        v2f a1f = *(const v2f*)(s_rel + (16 + l16) * LDA + 4 * kb + 2 * half);
        #pragma unroll
        for (int nt = 0; nt < 4; ++nt) {
            v2f b = *(const v2f*)(sm + SM_QT + (16 * nt + l16) * LDA + 4 * kb + 2 * half);
            att[0][nt] = wmma4(a0f, b, att[0][nt]);
            att[1][nt] = wmma4(a1f, b, att[1][nt]);
        }
    }

    // ---- row softmax over L=64; write w into stage as A-layout [32][64] ----
    #pragma unroll
    for (int t = 0; t < 2; ++t) {
        #pragma unroll
        for (int v = 0; v < 8; ++v) {
            const int m = 16 * t + v + 8 * half;
            float mx = fmaxf(fmaxf(att[t][0][v], att[t][1][v]),
                             fmaxf(att[t][2][v], att[t][3][v]));
            mx = half16_reduce_max(mx);
            float e0 = __expf(att[t][0][v] - mx), e1 = __expf(att[t][1][v] - mx);
            float e2 = __expf(att[t][2][v] - mx), e3 = __expf(att[t][3][v] - mx);
            float sum = half16_reduce_sum(e0 + e1 + e2 + e3);
            float inv = 1.0f / sum;
            s_stg[m * LDA + l16 +  0] = e0 * inv;
            s_stg[m * LDA + l16 + 16] = e1 * inv;
            s_stg[m * LDA + l16 + 32] = e2 * inv;
            s_stg[m * LDA + l16 + 48] = e3 * inv;
        }
    }
    lds_wave_sync();

    // ---- GEMM2: q[32x128] = w[32x64] * Q  (B[k=l][n=d] = Q[l][d], strided) -
    v8f acc[2][8] = {};
    for (int kb = 0; kb < 16; ++kb) {
        v2f a0f = *(const v2f*)(s_stg + (     l16) * LDA + 4 * kb + 2 * half);
        v2f a1f = *(const v2f*)(s_stg + (16 + l16) * LDA + 4 * kb + 2 * half);
        const int k0 = 4 * kb + 2 * half;
        #pragma unroll
        for (int nt = 0; nt < 8; ++nt) {
            const int n = 16 * nt + l16;
            v2f b;
            b.x = sm[SM_QT + (k0 + 0) * LDA + n];
            b.y = sm[SM_QT + (k0 + 1) * LDA + n];
            acc[0][nt] = wmma4(a0f, b, acc[0][nt]);
            acc[1][nt] = wmma4(a1f, b, acc[1][nt]);
        }
    }
    lds_wave_sync();
    #pragma unroll
    for (int t = 0; t < 2; ++t)
        #pragma unroll
        for (int nt = 0; nt < 8; ++nt)
            #pragma unroll
            for (int v = 0; v < 8; ++v)
                s_stg[(16 * t + v + 8 * half) * LDA + 16 * nt + l16] = acc[t][nt][v];
    lds_wave_sync();

    // ---- GEMM3: h = relu(q * W1q^T + consth) -------------------------------
    #pragma unroll
    for (int t = 0; t < 2; ++t)
        #pragma unroll
        for (int nt = 0; nt < 8; ++nt) acc[t][nt] = (v8f){};
    for (int kb = 0; kb < 32; ++kb) {
        v2f a0f = *(const v2f*)(s_stg + (     l16) * LDA + 4 * kb + 2 * half);
        v2f a1f = *(const v2f*)(s_stg + (16 + l16) * LDA + 4 * kb + 2 * half);
        #pragma unroll
        for (int nt = 0; nt < 8; ++nt) {
            v2f b = *(const v2f*)(sm + SM_W1Q + (16 * nt + l16) * LDA + 4 * kb + 2 * half);
            acc[0][nt] = wmma4(a0f, b, acc[0][nt]);
            acc[1][nt] = wmma4(a1f, b, acc[1][nt]);
        }
    }
    lds_wave_sync();
    #pragma unroll
    for (int t = 0; t < 2; ++t)
        #pragma unroll
        for (int nt = 0; nt < 8; ++nt) {
            const float ch = sm[SM_CH + 16 * nt + l16];
            #pragma unroll
            for (int v = 0; v < 8; ++v)
                s_stg[(16 * t + v + 8 * half) * LDA + 16 * nt + l16] =
                    fmaxf(acc[t][nt][v] + ch, 0.0f);
        }
    lds_wave_sync();

    // ---- GEMM4: out = h * W2^T + b2 ----------------------------------------
    #pragma unroll
    for (int t = 0; t < 2; ++t)
        #pragma unroll
        for (int nt = 0; nt < 8; ++nt) acc[t][nt] = (v8f){};
    for (int kb = 0; kb < 32; ++kb) {
        v2f a0f = *(const v2f*)(s_stg + (     l16) * LDA + 4 * kb + 2 * half);
        v2f a1f = *(const v2f*)(s_stg + (16 + l16) * LDA + 4 * kb + 2 * half);
        #pragma unroll
        for (int nt = 0; nt < 8; ++nt) {
            v2f b = *(const v2f*)(sm + SM_W2 + (16 * nt + l16) * LDA + 4 * kb + 2 * half);
            acc[0][nt] = wmma4(a0f, b, acc[0][nt]);
            acc[1][nt] = wmma4(a1f, b, acc[1][nt]);
        }
    }

    // ---- scores[m] = sum_o rel[m][o] * (out[m][o] + b2[o]) -----------------
    #pragma unroll
    for (int t = 0; t < 2; ++t) {
        #pragma unroll
        for (int v = 0; v < 8; ++v) {
            const int m = 16 * t + v + 8 * half;
            float p = 0.0f;
            #pragma unroll
            for (int nt = 0; nt < 8; ++nt) {
                const int o = 16 * nt + l16;
                p += s_rel[m * LDA + o] * (acc[t][nt][v] + sm[SM_B2 + o]);
            }
            p = half16_reduce_sum(p);
            if (l16 == 0) scores[a0 + m] = p;
        }
    }
}

// ---------------------------------------------------------------------------
// Global softmax over A: pass 1 (max, sum) then pass 2 (normalize).
// ---------------------------------------------------------------------------
__global__ __launch_bounds__(1024)
void softmax_stats_kernel(const float* __restrict__ scores, float* __restrict__ stats) {
    __shared__ float red[1024];
    const int tid = threadIdx.x;
    float m = -3.402823466e38f;
    for (int i = tid; i < A_N; i += 1024) m = fmaxf(m, scores[i]);
    red[tid] = m; __syncthreads();
    for (int s = 512; s > 0; s >>= 1) {
        if (tid < s) red[tid] = fmaxf(red[tid], red[tid + s]);
        __syncthreads();
    }
    const float gmax = red[0]; __syncthreads();
    float sum = 0.0f;
    for (int i = tid; i < A_N; i += 1024) sum += __expf(scores[i] - gmax);
    red[tid] = sum; __syncthreads();
    for (int s = 512; s > 0; s >>= 1) {
        if (tid < s) red[tid] += red[tid + s];
        __syncthreads();
    }
    if (tid == 0) { stats[0] = gmax; stats[1] = red[0]; }
}

__global__ void softmax_norm_kernel(const float* __restrict__ scores,
                                    const float* __restrict__ stats,
                                    float* __restrict__ out) {
    const int i = blockIdx.x * blockDim.x + threadIdx.x;
    if (i < A_N) out[i] = __expf(scores[i] - stats[0]) * (1.0f / stats[1]);
}

// ---------------------------------------------------------------------------
extern "C" void kernel_launch(void* const* d_in, const int* in_sizes, int n_in,
                              void* d_out, int out_size, void* d_ws, size_t ws_size,
                              hipStream_t stream) {
    const int*   action_ids = (const int*)  d_in[0];
    const float* emb_table  = (const float*)d_in[1];
    const float* question_t = (const float*)d_in[2];
    const float* history_t  = (const float*)d_in[3];
    const float* W1         = (const float*)d_in[4];
    const float* b1         = (const float*)d_in[5];
    const float* W2         = (const float*)d_in[6];
    const float* b2         = (const float*)d_in[7];
    float* out = (float*)d_out;

    float* ws     = (float*)d_ws;
    float* consth = ws;                // 128 floats
    float* scores = ws + 128;          // A_N floats
    float* stats  = ws + 128 + A_N;    // 2 floats

    (void)hipFuncSetAttribute((const void*)policy_scores_kernel,
                              hipFuncAttributeMaxDynamicSharedMemorySize, SM_BYTES);

    consth_kernel<<<1, 128, 0, stream>>>(history_t, W1, b1, consth);

    const int blocks = (TILES + WAVES - 1) / WAVES;   // 782
    policy_scores_kernel<<<blocks, 128, SM_BYTES, stream>>>(
        action_ids, emb_table, question_t, W1, W2, b2, consth, scores);

    softmax_stats_kernel<<<1, 1024, 0, stream>>>(scores, stats);
    softmax_norm_kernel<<<(A_N + 255) / 256, 256, 0, stream>>>(scores, stats, out);
}